// MedianPool1d_34222299415228
// MI455X (gfx1250) — compile-verified
//
#include <hip/hip_runtime.h>

// MedianPool1d, K=4, stride=1, reflect pad (1,2), lower median (sorted[1]).
// x: [L, C] f32, out: [L, C] f32.  L = 524288, C = 64.
//
// Memory-bound kernel: 268 MB of traffic -> ~11.5us floor @ 23.3 TB/s.
// Strategy: per block, async-DMA a (ROWS+3)-row halo tile from global into LDS
// (CDNA5 GLOBAL_LOAD_ASYNC_TO_LDS_B128, tracked by ASYNCcnt), wait + barrier,
// then each thread computes one float4 of output from 4 LDS rows and writes it
// with a non-temporal 128-bit store.

#define C_CH      64
#define CG        16          // float4 groups per row (64 floats / 4)
#define ROWS      16          // output rows per block
#define TILE_ROWS (ROWS + 3)  // halo rows: -1 .. +2  (19)
#define THREADS   256

typedef float v4f __attribute__((ext_vector_type(4)));

__device__ __forceinline__ float med4_lo(float a, float b, float c, float d) {
    // second-smallest of 4 (== sorted[1])
    float lo1 = fminf(a, b), hi1 = fmaxf(a, b);
    float lo2 = fminf(c, d), hi2 = fmaxf(c, d);
    return fminf(fmaxf(lo1, lo2), fminf(hi1, hi2));
}

__global__ __launch_bounds__(THREADS) void median_pool1d_kernel(
    const float* __restrict__ x, float* __restrict__ out, int L)
{
    __shared__ __align__(16) float tile[TILE_ROWS * C_CH]; // 4864 B

    const int tid        = threadIdx.x;
    const int block_row0 = blockIdx.x * ROWS;

    // ---- Stage halo tile: rows (block_row0-1 .. block_row0+ROWS+1), reflected,
    //      via CDNA5 async global->LDS DMA (ASYNCcnt).
    const unsigned ldsBase = (unsigned)(size_t)(&tile[0]); // low 32 bits = LDS offset
    for (int t = tid; t < TILE_ROWS * CG; t += THREADS) {
        const int j  = t >> 4;       // tile row 0..18
        const int cg = t & 15;       // float4 group within row
        int g = block_row0 - 1 + j;  // global row, reflect at both ends
        if (g < 0)  g = -g;                // map(-1) -> 1
        if (g >= L) g = 2 * L - 2 - g;     // map(L) -> L-2, map(L+1) -> L-3
        const unsigned gOff   = (unsigned)g * (C_CH * 4u) + (unsigned)cg * 16u;
        const unsigned ldsOff = ldsBase + (unsigned)t * 16u;
        asm volatile("global_load_async_to_lds_b128 %0, %1, %2"
                     :
                     : "v"(ldsOff), "v"(gOff), "s"(x)
                     : "memory");
    }
    asm volatile("s_wait_asynccnt 0" ::: "memory");
    __syncthreads();

    // ---- Compute: output row (block_row0 + r) uses LDS rows r..r+3.
    const int r    = tid >> 4;   // 0..15
    const int cg   = tid & 15;   // 0..15
    const int orow = block_row0 + r;

    const v4f* lp = (const v4f*)(&tile[0]);
    v4f a = lp[(r + 0) * CG + cg];
    v4f b = lp[(r + 1) * CG + cg];
    v4f c = lp[(r + 2) * CG + cg];
    v4f d = lp[(r + 3) * CG + cg];

    v4f m;
    m.x = med4_lo(a.x, b.x, c.x, d.x);
    m.y = med4_lo(a.y, b.y, c.y, d.y);
    m.z = med4_lo(a.z, b.z, c.z, d.z);
    m.w = med4_lo(a.w, b.w, c.w, d.w);

    if (orow < L) {
        v4f* op = (v4f*)(out + (size_t)orow * C_CH + (size_t)cg * 4);
        __builtin_nontemporal_store(m, op);  // write-once output: keep out of L2
    }
}

extern "C" void kernel_launch(void* const* d_in, const int* in_sizes, int n_in,
                              void* d_out, int out_size, void* d_ws, size_t ws_size,
                              hipStream_t stream) {
    const float* x = (const float*)d_in[0];
    float* out     = (float*)d_out;
    const int L    = in_sizes[0] / C_CH;   // 524288
    const int grid = (L + ROWS - 1) / ROWS;
    median_pool1d_kernel<<<grid, THREADS, 0, stream>>>(x, out, L);
}